// VQVAEZMultiScale_20890720928600
// MI455X (gfx1250) — compile-verified
//
#include <hip/hip_runtime.h>
#include <hip/hip_bf16.h>
#include <math.h>

typedef float v2f __attribute__((ext_vector_type(2)));
typedef float v4f __attribute__((ext_vector_type(4)));
typedef float v8f __attribute__((ext_vector_type(8)));

#define NCODES   2048
#define CDIM     64
#define NPOS     32768            // 8 * 64 * 64
#define X0_N     2097152          // 8*64*64*64
#define D2_N     524288           // 8*32*32*64
#define D4_N     131072           // 8*16*16*64
#define ZIDX_OFF 2097152
#define ZIDX_N   65536            // 8*2*64*64
#define QUANT_OFF (ZIDX_OFF + ZIDX_N)
#define BPAD     68               // LDS row stride (floats): 16B-aligned, banks = 4*code + off

// ---------------- transpose (b,c,h,w) -> (b,h,w,c) ----------------
__global__ __launch_bounds__(256) void k_transpose(const float* __restrict__ in,
                                                   float* __restrict__ x0) {
  __shared__ float tile[64][65];
  const int b = blockIdx.x >> 6;
  const int y = blockIdx.x & 63;
  {
    const int x = threadIdx.x & 63;
    const int cq = threadIdx.x >> 6;        // 0..3
    #pragma unroll
    for (int i = 0; i < 16; ++i) {
      const int c = cq * 16 + i;
      tile[c][x] = in[(((size_t)b * 64 + c) * 64 + y) * 64 + x];
    }
  }
  __syncthreads();
  {
    const int c = threadIdx.x & 63;
    const int xq = threadIdx.x >> 6;
    #pragma unroll
    for (int i = 0; i < 16; ++i) {
      const int xx = xq * 16 + i;
      x0[(((size_t)b * 64 + y) * 64 + xx) * 64 + c] = tile[c][xx];
    }
  }
}

// ---------------- bilinear(antialias) downsample x2 : x0 -> (8,32,32,64) ----
__global__ __launch_bounds__(256) void k_down2(const float* __restrict__ x0,
                                               float* __restrict__ dn2) {
  const int gid = blockIdx.x * 256 + threadIdx.x;   // 524288 total
  const int c  = gid & 63;
  const int xo = (gid >> 6) & 31;
  const int yo = (gid >> 11) & 31;
  const int b  = gid >> 16;
  const float w4[4] = {0.125f, 0.375f, 0.375f, 0.125f};
  float s = 0.0f;
  #pragma unroll
  for (int j = 0; j < 4; ++j) {
    int yi = 2 * yo - 1 + j; yi = yi < 0 ? 0 : (yi > 63 ? 63 : yi);
    #pragma unroll
    for (int i = 0; i < 4; ++i) {
      int xi = 2 * xo - 1 + i; xi = xi < 0 ? 0 : (xi > 63 ? 63 : xi);
      s += w4[j] * w4[i] * x0[(((size_t)b * 64 + yi) * 64 + xi) * 64 + c];
    }
  }
  dn2[gid] = s;
}

// ---------------- bilinear(antialias) downsample x4 : x0 -> (8,16,16,64) ----
__global__ __launch_bounds__(256) void k_down4(const float* __restrict__ x0,
                                               float* __restrict__ dn4) {
  const int gid = blockIdx.x * 256 + threadIdx.x;   // 131072 total
  const int c  = gid & 63;
  const int xo = (gid >> 6) & 15;
  const int yo = (gid >> 10) & 15;
  const int b  = gid >> 14;
  const float w8[8] = {0.03125f, 0.09375f, 0.15625f, 0.21875f,
                       0.21875f, 0.15625f, 0.09375f, 0.03125f};
  float s = 0.0f;
  #pragma unroll
  for (int j = 0; j < 8; ++j) {
    int yi = 4 * yo - 2 + j; yi = yi < 0 ? 0 : (yi > 63 ? 63 : yi);
    #pragma unroll
    for (int i = 0; i < 8; ++i) {
      int xi = 4 * xo - 2 + i; xi = xi < 0 ? 0 : (xi > 63 ? 63 : xi);
      s += w8[j] * w8[i] * x0[(((size_t)b * 64 + yi) * 64 + xi) * 64 + c];
    }
  }
  dn4[gid] = s;
}

// ---------------- codebook squared norms (cb0, cb1) ----------------
__global__ __launch_bounds__(256) void k_cbnorm(const float* __restrict__ codebooks,
                                                float* __restrict__ ccn) {
  const int gid = blockIdx.x * 256 + threadIdx.x;   // 4096 total
  const int cbi = gid >> 11;                        // 0 -> cb0, 1 -> cb1
  const int k   = gid & 2047;
  const float* row = codebooks + ((size_t)cbi * NCODES + k) * CDIM;
  float s = 0.0f;
  #pragma unroll 8
  for (int c = 0; c < CDIM; ++c) { float v = row[c]; s += v * v; }
  ccn[gid] = s;
}

// ---------------- WMMA VQ match ----------------
// grid = 1024 blocks x 256 threads. seg = blockIdx>>8:
//   seg 0: x0 candidates vs cb0        seg 1: up(dn2) vs cb0
//   seg 2: up(dn4) vs cb0              seg 3: x0 vs cb1 (argmin only)
// Tracks per row: min score, argmin idx, sum = sum_k exp(min - score_k).
__global__ __launch_bounds__(256) void k_match(
    const float* __restrict__ x0, const float* __restrict__ dn2,
    const float* __restrict__ dn4, const float* __restrict__ codebooks,
    const float* __restrict__ ccn,
    int* __restrict__ out_idx, float* __restrict__ out_sum) {
  __shared__ float ldsB[2][16 * BPAD];
  __shared__ float ldsCC[NCODES];

  const int seg = blockIdx.x >> 8;
  const int wgInSeg = blockIdx.x & 255;
  const int wave = threadIdx.x >> 5;
  const int lane = threadIdx.x & 31;
  const int col  = lane & 15;
  const int hi   = lane >> 4;
  const int rowbase = wgInSeg * 128 + wave * 16;
  const int myrow = rowbase + col;

  const float* cb = codebooks + (size_t)((seg < 3) ? 0 : 1) * NCODES * CDIM;
  const float* cc = ccn + ((seg < 3) ? 0 : NCODES);

  // stage all 2048 codebook norms once
  #pragma unroll
  for (int r = 0; r < 8; ++r)
    ldsCC[threadIdx.x + r * 256] = cc[threadIdx.x + r * 256];

  // per-lane A-row pointer
  const float* qrow;
  {
    const int b = myrow >> 12;
    const int rem = myrow & 4095;
    const int y = rem >> 6;
    const int x = rem & 63;
    if (seg == 1)       qrow = dn2 + (size_t)(((b << 5) + (y >> 1)) * 32 + (x >> 1)) * CDIM;
    else if (seg == 2)  qrow = dn4 + (size_t)(((b << 4) + (y >> 2)) * 16 + (x >> 2)) * CDIM;
    else                qrow = x0 + (size_t)myrow * CDIM;   // seg 0 and 3
  }

  // A fragments: lane holds row (lane&15), f32 pair at channel 4*kk + 2*hi
  v2f afrag[16];
  #pragma unroll
  for (int kk = 0; kk < 16; ++kk)
    afrag[kk] = *reinterpret_cast<const v2f*>(qrow + kk * 4 + hi * 2);

  // B staging geometry: one b128 per thread per tile
  const int scode = threadIdx.x >> 4;          // 0..15
  const int sch   = (threadIdx.x & 15) * 4;    // 0..60
  *reinterpret_cast<v4f*>(&ldsB[0][scode * BPAD + sch]) =
      *reinterpret_cast<const v4f*>(&cb[(size_t)scode * CDIM + sch]);

  float minv[8], sumv[8];
  int idxv[8];
  #pragma unroll
  for (int v = 0; v < 8; ++v) { minv[v] = 3.0e38f; sumv[v] = 0.0f; idxv[v] = 0; }

  for (int t = 0; t < 128; ++t) {
    __syncthreads();      // ldsB[t&1] staged for everyone; prev compute done

    v4f breg;
    if (t < 127)          // prefetch next tile while WMMAs run
      breg = *reinterpret_cast<const v4f*>(
          &cb[((size_t)(t + 1) * 16 + scode) * CDIM + sch]);

    const float* bbuf = ldsB[t & 1];
    v8f acc = {0.f, 0.f, 0.f, 0.f, 0.f, 0.f, 0.f, 0.f};
    #pragma unroll
    for (int kk = 0; kk < 16; ++kk) {
      v2f bfrag = *reinterpret_cast<const v2f*>(&bbuf[col * BPAD + kk * 4 + hi * 2]);
      acc = __builtin_amdgcn_wmma_f32_16x16x4_f32(
          false, afrag[kk], false, bfrag, (short)0, acc, false, false);
    }

    const int n = t * 16 + col;
    const float cn = ldsCC[n];
    #pragma unroll
    for (int v = 0; v < 8; ++v) {
      const float sc = fmaf(-2.0f, acc[v], cn);
      const bool take = sc < minv[v];
      const float e = __expf(-fabsf(sc - minv[v]));
      const float mul = take ? e : 1.0f;
      const float add = take ? 1.0f : e;
      sumv[v] = fmaf(sumv[v], mul, add);
      idxv[v] = take ? n : idxv[v];
      minv[v] = take ? sc : minv[v];
    }

    if (t < 127)
      *reinterpret_cast<v4f*>(&ldsB[(t + 1) & 1][scode * BPAD + sch]) = breg;
  }

  // merge 16 column-partials per row (xor within 16-lane half)
  #pragma unroll
  for (int m = 1; m <= 8; m <<= 1) {
    #pragma unroll
    for (int v = 0; v < 8; ++v) {
      const float om = __shfl_xor(minv[v], m, 32);
      const int   oi = __shfl_xor(idxv[v], m, 32);
      const float os = __shfl_xor(sumv[v], m, 32);
      const float nm = fminf(minv[v], om);
      const float ns = fmaf(sumv[v], __expf(nm - minv[v]), os * __expf(nm - om));
      if (om < minv[v] || (om == minv[v] && oi < idxv[v])) idxv[v] = oi;
      minv[v] = nm;
      sumv[v] = ns;
    }
  }

  if (col == 0) {   // lane 0 writes rows 0..7, lane 16 writes rows 8..15
    #pragma unroll
    for (int v = 0; v < 8; ++v) {
      const int row = rowbase + v + hi * 8;
      const int o = seg * NPOS + row;
      out_idx[o] = idxv[v];
      out_sum[o] = sumv[v];
    }
  }
}

// ---------------- finalize: pick scale, gather codebooks, write outputs -----
__global__ __launch_bounds__(256) void k_finalize(
    const float* __restrict__ codebooks, const int* __restrict__ out_idx,
    const float* __restrict__ out_sum, float* __restrict__ out) {
  const int gid = blockIdx.x * 256 + threadIdx.x;   // 2097152 total
  const int x  = gid & 63;
  const int y  = (gid >> 6) & 63;
  const int ch = (gid >> 12) & 63;
  const int b  = gid >> 18;
  const int n  = (b << 12) | (y << 6) | x;

  // argmax over scales of zsoft_max = 1/sum  == argmin of sum (first max wins)
  const float s0 = out_sum[n];
  const float s1 = out_sum[NPOS + n];
  const float s2 = out_sum[2 * NPOS + n];
  int sstar = 0;
  float sbest = s0;
  if (s1 < sbest) { sbest = s1; sstar = 1; }
  if (s2 < sbest) { sbest = s2; sstar = 2; }

  const int zidx1 = out_idx[sstar * NPOS + n];
  const int zidx2 = out_idx[3 * NPOS + n];
  const float q = 0.5f * (codebooks[(size_t)zidx1 * CDIM + ch] +
                          codebooks[((size_t)NCODES + zidx2) * CDIM + ch]);
  out[QUANT_OFF + (((size_t)b * 64 + ch) * 64 + y) * 64 + x] = q;
  if (ch == 0) {
    out[ZIDX_OFF + (size_t)b * 8192 + y * 64 + x] = (float)zidx1;
    out[ZIDX_OFF + (size_t)b * 8192 + 4096 + y * 64 + x] = (float)zidx2;
  }
}

extern "C" void kernel_launch(void* const* d_in, const int* in_sizes, int n_in,
                              void* d_out, int out_size, void* d_ws, size_t ws_size,
                              hipStream_t stream) {
  const float* input = (const float*)d_in[0];       // (8,64,64,64) bchw
  const float* codebooks = (const float*)d_in[1];   // (4,2048,64)
  float* out = (float*)d_out;

  // workspace layout (floats)
  float* ws   = (float*)d_ws;
  float* x0   = ws;                       // 2097152
  float* dn2  = x0 + X0_N;                // 524288
  float* dn4  = dn2 + D2_N;               // 131072
  float* ccn  = dn4 + D4_N;               // 4096
  float* sums = ccn + 4096;               // 4*32768
  int*   idxs = (int*)(sums + 4 * NPOS);  // 4*32768

  k_transpose<<<512, 256, 0, stream>>>(input, x0);
  k_down2<<<D2_N / 256, 256, 0, stream>>>(x0, dn2);
  k_down4<<<D4_N / 256, 256, 0, stream>>>(x0, dn4);
  k_cbnorm<<<4096 / 256, 256, 0, stream>>>(codebooks, ccn);
  k_match<<<1024, 256, 0, stream>>>(x0, dn2, dn4, codebooks, ccn, idxs, sums);
  // encode0 = input, unchanged
  hipMemcpyAsync(out, input, (size_t)X0_N * sizeof(float),
                 hipMemcpyDeviceToDevice, stream);
  k_finalize<<<X0_N / 256, 256, 0, stream>>>(codebooks, idxs, sums, out);
}